// DirectedHyperConvLayer_26070451486831
// MI455X (gfx1250) — compile-verified
//
#include <hip/hip_runtime.h>
#include <hip/hip_bf16.h>
#include <cstdint>

// ---------------------------------------------------------------------------
// DirectedHyperConvLayer on MI455X (gfx1250):
//   msg_tar = COO(tar) @ pois_embs                         [50000 x 128]
//   out     = dropout(relu(COO(src) @ msg_tar), p=0.3)     [100000 x 128]
//
// Atomic-throughput analysis: direct COO scatter needs 410M f32 atomics
// (~ms at L2 atomic rates) vs a ~13us HBM floor. So: build CSR on device
// (hist + scan + scatter, ~6.4M int atomics), then gather-based wave-per-row
// SpMM with ZERO atomics in the hot loop. CDNA5 paths: double-buffered
// async-to-LDS staging of CSR (col,val) chunks with partial ASYNCcnt waits,
// global_prefetch of gathered rows, b128 gathers/stores, ds_load_b64
// broadcasts. ReLU+dropout fused into SpMM2's store.
// WMMA does not apply: unstructured sparsity has no shared-B 16x16 tiles.
// ---------------------------------------------------------------------------

#define WAVES_PER_BLOCK 8
#define WAVE 32

#if defined(__gfx1250__) && __has_builtin(__builtin_amdgcn_global_load_async_to_lds_b32)
#define HAVE_ASYNC_LDS 1
#else
#define HAVE_ASYNC_LDS 0
#endif

#if HAVE_ASYNC_LDS
typedef __attribute__((address_space(1))) int g1int;
typedef __attribute__((address_space(3))) int l3int;

__device__ __forceinline__ void async_ld_b32(const void* gsrc, void* ldst) {
  __builtin_amdgcn_global_load_async_to_lds_b32(
      (g1int*)(gsrc), (l3int*)(ldst), /*offset=*/0, /*cpol=*/0);
}
__device__ __forceinline__ void wait_async0() {
#if __has_builtin(__builtin_amdgcn_s_wait_asynccnt)
  __builtin_amdgcn_s_wait_asynccnt(0);
#else
  asm volatile("s_wait_asynccnt 0x0" ::: "memory");
#endif
}
__device__ __forceinline__ void wait_async2() {   // oldest chunk done (in-order)
#if __has_builtin(__builtin_amdgcn_s_wait_asynccnt)
  __builtin_amdgcn_s_wait_asynccnt(2);
#else
  asm volatile("s_wait_asynccnt 0x2" ::: "memory");
#endif
}
#endif

__device__ __forceinline__ uint32_t mix_hash(uint32_t x) {
  x ^= 42u * 0x9E3779B9u;       // dropout key
  x *= 0x85EBCA6Bu;  x ^= x >> 13;
  x *= 0xC2B2AE35u;  x ^= x >> 16;
  return x;
}

// ---------------------------------------------------------------------------
// CSR gather SpMM: one wave per output row. Lane l owns dims [4l, 4l+4).
// (col,val) chunks of 32 are staged into double-buffered LDS via async-to-LDS;
// chunk i+1 is issued before consuming chunk i, with s_wait_asynccnt 2 relying
// on in-order async completion. Inner loop: one ds_load_b64 broadcast + one
// b128 gather + packed FMAs per nnz. One b128 store per row at the end;
// final_mode fuses relu + inverted dropout.
// ---------------------------------------------------------------------------
__global__ __launch_bounds__(256) void spmm_csr_kernel(
    const int* __restrict__ rowptr,   // [nrows+1]
    const int* __restrict__ csr_col,  // [nnz]
    const float* __restrict__ csr_val,// [nnz]
    const float4* __restrict__ X4,    // dense [*, 128] as [*, 32] float4
    float* __restrict__ Y,            // [nrows, 128]
    int nrows, int final_mode)
{
  __shared__ int2 s_cv[2][WAVES_PER_BLOCK][WAVE];   // .x = col, .y = val bits

  const int lane = threadIdx.x & (WAVE - 1);
  const int wid  = threadIdx.x >> 5;
  const int row  = blockIdx.x * WAVES_PER_BLOCK + wid;
  if (row >= nrows) return;

  const int start = rowptr[row];
  const int end   = rowptr[row + 1];

  float4 acc = {0.f, 0.f, 0.f, 0.f};

  if (start < end) {
    // Stage first chunk.
    {
      const int m0   = (end - start < WAVE) ? (end - start) : WAVE;
      const int idx0 = start + ((lane < m0) ? lane : (m0 - 1));  // clamp tail
#if HAVE_ASYNC_LDS
      async_ld_b32(csr_col + idx0, &s_cv[0][wid][lane].x);
      async_ld_b32(csr_val + idx0, &s_cv[0][wid][lane].y);
#else
      s_cv[0][wid][lane].x = csr_col[idx0];
      s_cv[0][wid][lane].y = __float_as_int(csr_val[idx0]);
#endif
    }

    int buf = 0;
    for (int chunk = start; chunk < end; chunk += WAVE, buf ^= 1) {
      const int next = chunk + WAVE;
      const bool has_next = next < end;           // wave-uniform
      if (has_next) {                             // stage chunk i+1 into other buf
        const int mn   = (end - next < WAVE) ? (end - next) : WAVE;
        const int idxn = next + ((lane < mn) ? lane : (mn - 1));
#if HAVE_ASYNC_LDS
        async_ld_b32(csr_col + idxn, &s_cv[buf ^ 1][wid][lane].x);
        async_ld_b32(csr_val + idxn, &s_cv[buf ^ 1][wid][lane].y);
#else
        s_cv[buf ^ 1][wid][lane].x = csr_col[idxn];
        s_cv[buf ^ 1][wid][lane].y = __float_as_int(csr_val[idxn]);
#endif
      }
#if HAVE_ASYNC_LDS
      if (has_next) wait_async2(); else wait_async0();
#endif

      const int m = (end - chunk < WAVE) ? (end - chunk) : WAVE;

      // Prefetch rows we are about to gather (512B each; grab two lines).
      if (lane < m) {
        const char* prow = (const char*)(X4 + (size_t)s_cv[buf][wid][lane].x * 32);
        __builtin_prefetch(prow, 0, 1);
        __builtin_prefetch(prow + 256, 0, 1);
      }

      for (int k = 0; k < m; ++k) {
        const int2  cv = s_cv[buf][wid][k];       // ds_load_b64 broadcast
        const float v  = __int_as_float(cv.y);
        const float4 x = X4[(size_t)cv.x * 32 + lane];   // global_load_b128
        acc.x += v * x.x;  acc.y += v * x.y;
        acc.z += v * x.z;  acc.w += v * x.w;
      }
    }
  }

  if (final_mode) {
    const uint32_t thresh = 3006477107u;  // floor(0.7 * 2^32): keep-prob 0.7
    const uint32_t g = (uint32_t)row * 128u + (uint32_t)lane * 4u;
    acc.x = (mix_hash(g + 0) < thresh) ? fmaxf(acc.x, 0.f) * (1.0f / 0.7f) : 0.f;
    acc.y = (mix_hash(g + 1) < thresh) ? fmaxf(acc.y, 0.f) * (1.0f / 0.7f) : 0.f;
    acc.z = (mix_hash(g + 2) < thresh) ? fmaxf(acc.z, 0.f) * (1.0f / 0.7f) : 0.f;
    acc.w = (mix_hash(g + 3) < thresh) ? fmaxf(acc.w, 0.f) * (1.0f / 0.7f) : 0.f;
  }
  ((float4*)(Y + (size_t)row * 128))[lane] = acc;   // global_store_b128
}

// ------------------------- CSR construction kernels ------------------------

__global__ __launch_bounds__(256) void zero_i32_kernel(int* __restrict__ p, int n) {
  int i = blockIdx.x * blockDim.x + threadIdx.x;
  const int stride = gridDim.x * blockDim.x;
  for (; i < n; i += stride) p[i] = 0;
}

__global__ __launch_bounds__(256) void hist_kernel(
    const int* __restrict__ rows, int nnz, int* __restrict__ cnt) {
  int i = blockIdx.x * blockDim.x + threadIdx.x;
  const int stride = gridDim.x * blockDim.x;
  for (; i < nnz; i += stride)
    __hip_atomic_fetch_add(&cnt[rows[i]], 1, __ATOMIC_RELAXED, __HIP_MEMORY_SCOPE_AGENT);
}

// Single-workgroup exclusive scan: rowptr[i] = sum(cnt[0..i)), rowptr[n]=total;
// also initializes cursor[i] = rowptr[i] for the scatter pass.
__global__ __launch_bounds__(1024) void scan_kernel(
    const int* __restrict__ cnt, int n,
    int* __restrict__ rowptr, int* __restrict__ cursor) {
  __shared__ int sdata[1024];
  const int tid = threadIdx.x;
  int running = 0;
  for (int base = 0; base < n; base += 1024) {
    const int i = base + tid;
    const int x = (i < n) ? cnt[i] : 0;
    sdata[tid] = x;
    __syncthreads();
    for (int off = 1; off < 1024; off <<= 1) {
      const int t = (tid >= off) ? sdata[tid - off] : 0;
      __syncthreads();
      sdata[tid] += t;
      __syncthreads();
    }
    const int excl = sdata[tid] - x;
    if (i < n) { rowptr[i] = running + excl; cursor[i] = running + excl; }
    const int tot = sdata[1023];
    __syncthreads();          // protect sdata[1023] before next iter's write
    running += tot;
  }
  if (tid == 0) rowptr[n] = running;
}

__global__ __launch_bounds__(256) void scatter_kernel(
    const int* __restrict__ rows, const int* __restrict__ cols,
    const float* __restrict__ vals, int nnz,
    int* __restrict__ cursor, int* __restrict__ ccol, float* __restrict__ cval) {
  int i = blockIdx.x * blockDim.x + threadIdx.x;
  const int stride = gridDim.x * blockDim.x;
  for (; i < nnz; i += stride) {
    const int r = rows[i];
    const int p = __hip_atomic_fetch_add(&cursor[r], 1, __ATOMIC_RELAXED,
                                         __HIP_MEMORY_SCOPE_AGENT);
    ccol[p] = cols[i];
    cval[p] = vals[i];
  }
}

// ---------------- Fallback path (small ws): COO atomic scatter -------------

__global__ __launch_bounds__(256) void zero_f4_kernel(float4* __restrict__ p, long n4) {
  long i = (long)blockIdx.x * blockDim.x + threadIdx.x;
  long stride = (long)gridDim.x * blockDim.x;
  float4 z = {0.f, 0.f, 0.f, 0.f};
  for (; i < n4; i += stride) p[i] = z;
}

__global__ __launch_bounds__(256) void spmm_scatter_kernel(
    const int* __restrict__ rows, const int* __restrict__ cols,
    const float* __restrict__ vals, int nnz,
    const float4* __restrict__ X4, float* __restrict__ Y) {
  __shared__ int   s_row[WAVES_PER_BLOCK][WAVE];
  __shared__ int   s_col[WAVES_PER_BLOCK][WAVE];
  __shared__ float s_val[WAVES_PER_BLOCK][WAVE];
  const int lane = threadIdx.x & (WAVE - 1);
  const int wid  = threadIdx.x >> 5;
  const int base = (blockIdx.x * WAVES_PER_BLOCK + wid) * WAVE;
  if (base >= nnz) return;
  const int m = (nnz - base < WAVE) ? (nnz - base) : WAVE;
  const int idx = base + ((lane < m) ? lane : (m - 1));
#if HAVE_ASYNC_LDS
  async_ld_b32(rows + idx, &s_row[wid][lane]);
  async_ld_b32(cols + idx, &s_col[wid][lane]);
  async_ld_b32(vals + idx, &s_val[wid][lane]);
  wait_async0();
#else
  s_row[wid][lane] = rows[idx];
  s_col[wid][lane] = cols[idx];
  s_val[wid][lane] = vals[idx];
#endif
  for (int k = 0; k < m; ++k) {
    const int   r = s_row[wid][k];
    const int   c = s_col[wid][k];
    const float v = s_val[wid][k];
    const float4 x = X4[(size_t)c * 32 + lane];
    float* y = Y + (size_t)r * 128 + lane * 4;
    __hip_atomic_fetch_add(y + 0, v * x.x, __ATOMIC_RELAXED, __HIP_MEMORY_SCOPE_AGENT);
    __hip_atomic_fetch_add(y + 1, v * x.y, __ATOMIC_RELAXED, __HIP_MEMORY_SCOPE_AGENT);
    __hip_atomic_fetch_add(y + 2, v * x.z, __ATOMIC_RELAXED, __HIP_MEMORY_SCOPE_AGENT);
    __hip_atomic_fetch_add(y + 3, v * x.w, __ATOMIC_RELAXED, __HIP_MEMORY_SCOPE_AGENT);
  }
}

__global__ __launch_bounds__(256) void finalize_kernel(float* __restrict__ out, int n) {
  int i = blockIdx.x * blockDim.x + threadIdx.x;
  const int stride = gridDim.x * blockDim.x;
  const uint32_t thresh = 3006477107u;
  for (; i < n; i += stride) {
    const float v = fmaxf(out[i], 0.f) * (1.0f / 0.7f);
    out[i] = (mix_hash((uint32_t)i) < thresh) ? v : 0.f;
  }
}

// ---------------------------------------------------------------------------

extern "C" void kernel_launch(void* const* d_in, const int* in_sizes, int n_in,
                              void* d_out, int out_size, void* d_ws, size_t ws_size,
                              hipStream_t stream) {
  const float* pois_embs = (const float*)d_in[0];
  const int*   tar_rows  = (const int*)d_in[1];
  const int*   tar_cols  = (const int*)d_in[2];
  const float* tar_vals  = (const float*)d_in[3];
  const int*   src_rows  = (const int*)d_in[4];
  const int*   src_cols  = (const int*)d_in[5];
  const float* src_vals  = (const float*)d_in[6];
  float* out = (float*)d_out;

  const int nnz     = in_sizes[1];     // 1.6M
  const int n_edges = 50000;
  const int n_pois  = 100000;

  // Workspace layout (4B elements), msg_tar first (16B aligned at base).
  const size_t e_msg = (size_t)n_edges * 128;
  size_t off = 0;
  float* msg_tar    = (float*)d_ws;              off += e_msg;
  int*   rowptr_tar = (int*)d_ws + off;          off += (size_t)n_edges + 1;
  int*   rowptr_src = (int*)d_ws + off;          off += (size_t)n_pois + 1;
  int*   cursor_tar = (int*)d_ws + off;          off += (size_t)n_edges;
  int*   cursor_src = (int*)d_ws + off;          off += (size_t)n_pois;
  int*   cnt_tar    = (int*)d_ws + off;          off += (size_t)n_edges;
  int*   cnt_src    = (int*)d_ws + off;          off += (size_t)n_pois;
  int*   tcol       = (int*)d_ws + off;          off += (size_t)nnz;
  float* tval       = (float*)d_ws + off;        off += (size_t)nnz;
  int*   scol       = (int*)d_ws + off;          off += (size_t)nnz;
  float* sval       = (float*)d_ws + off;        off += (size_t)nnz;
  const size_t need = off * 4;

  if (ws_size >= need) {
    // ---------------- CSR path: no atomics in the hot loops ----------------
    zero_i32_kernel<<<256, 256, 0, stream>>>(cnt_tar, n_edges);
    zero_i32_kernel<<<512, 256, 0, stream>>>(cnt_src, n_pois);
    hist_kernel<<<4096, 256, 0, stream>>>(tar_rows, nnz, cnt_tar);
    hist_kernel<<<4096, 256, 0, stream>>>(src_rows, nnz, cnt_src);
    scan_kernel<<<1, 1024, 0, stream>>>(cnt_tar, n_edges, rowptr_tar, cursor_tar);
    scan_kernel<<<1, 1024, 0, stream>>>(cnt_src, n_pois,  rowptr_src, cursor_src);
    scatter_kernel<<<4096, 256, 0, stream>>>(tar_rows, tar_cols, tar_vals, nnz,
                                             cursor_tar, tcol, tval);
    scatter_kernel<<<4096, 256, 0, stream>>>(src_rows, src_cols, src_vals, nnz,
                                             cursor_src, scol, sval);
    const int nblk1 = (n_edges + WAVES_PER_BLOCK - 1) / WAVES_PER_BLOCK;
    const int nblk2 = (n_pois  + WAVES_PER_BLOCK - 1) / WAVES_PER_BLOCK;
    spmm_csr_kernel<<<nblk1, 256, 0, stream>>>(rowptr_tar, tcol, tval,
        (const float4*)pois_embs, msg_tar, n_edges, /*final=*/0);
    spmm_csr_kernel<<<nblk2, 256, 0, stream>>>(rowptr_src, scol, sval,
        (const float4*)msg_tar, out, n_pois, /*final=*/1);
  } else {
    // ------------- Fallback: COO atomic scatter (needs only msg_tar) -------
    zero_f4_kernel<<<2048, 256, 0, stream>>>((float4*)msg_tar, (long)n_edges * 32);
    zero_f4_kernel<<<2048, 256, 0, stream>>>((float4*)out,     (long)n_pois * 32);
    const int nblk = (nnz + WAVES_PER_BLOCK * WAVE - 1) / (WAVES_PER_BLOCK * WAVE);
    spmm_scatter_kernel<<<nblk, 256, 0, stream>>>(
        tar_rows, tar_cols, tar_vals, nnz, (const float4*)pois_embs, msg_tar);
    spmm_scatter_kernel<<<nblk, 256, 0, stream>>>(
        src_rows, src_cols, src_vals, nnz, (const float4*)msg_tar, out);
    finalize_kernel<<<2048, 256, 0, stream>>>(out, n_pois * 128);
  }
}